// CRD_24945170055823
// MI455X (gfx1250) — compile-verified
//
#include <hip/hip_runtime.h>

typedef __attribute__((ext_vector_type(2))) float v2f;
typedef __attribute__((ext_vector_type(8))) float v8f;

#define NNODES 50000
#define NEDGES 800000
#define DIN    256
#define DOUT   64

// ---------------- init / degree kernels ----------------

__global__ void CRD_zero_out(float* __restrict__ out, int n) {
  int i = blockIdx.x * blockDim.x + threadIdx.x;
  if (i < n) out[i] = 0.0f;
}

__global__ void CRD_deg_init(float* __restrict__ deg, int n) {
  int i = blockIdx.x * blockDim.x + threadIdx.x;
  if (i < n) deg[i] = 1.0f;  // self-loop contribution
}

__global__ void CRD_deg_count(const int* __restrict__ dst, float* __restrict__ deg, int e) {
  int i = blockIdx.x * blockDim.x + threadIdx.x;
  if (i < e) atomicAdd(&deg[dst[i]], 1.0f);
}

__global__ void CRD_dinv(const float* __restrict__ deg, float* __restrict__ dinv, int n) {
  int i = blockIdx.x * blockDim.x + threadIdx.x;
  if (i < n) dinv[i] = rsqrtf(deg[i]);  // deg >= 1 always (self-loops)
}

// ---------------- WMMA GEMM: h = x @ W  (f32, 16x16x4) ----------------
// One wave computes a 16-row x 64-col tile: 4 accumulators (n-tiles of 16),
// 64 K-steps of 4 -> 256 v_wmma_f32_16x16x4_f32 per wave.
// A (16x4 f32) layout: lanes 0-15 hold M=lane, K={0,1}; lanes 16-31 K={2,3}.
// B (4x16 f32) layout: lanes 0-15 hold N=lane, K={0,1}; lanes 16-31 K={2,3}.
// C/D (16x16 f32): VGPR v, lanes 0-15 -> (M=v,   N=lane),
//                          lanes 16-31 -> (M=v+8, N=lane-16).

__global__ void CRD_gemm(const float* __restrict__ x, const float* __restrict__ W,
                         float* __restrict__ h) {
  const int wave = threadIdx.x >> 5;
  const int lane = threadIdx.x & 31;
  const int m0 = (blockIdx.x * 8 + wave) * 16;
  if (m0 >= NNODES) return;  // wave-uniform: EXEC stays all-ones for WMMA

  const int mrow = m0 + (lane & 15);
  const int ncol = lane & 15;
  const int koff = (lane < 16) ? 0 : 2;

  v8f acc0 = {}, acc1 = {}, acc2 = {}, acc3 = {};
  const float* xr = x + (size_t)mrow * DIN;

  for (int k0 = 0; k0 < DIN; k0 += 4) {
    v2f a;
    a.x = xr[k0 + koff];
    a.y = xr[k0 + koff + 1];

    const float* w0 = W + (size_t)(k0 + koff) * DOUT + ncol;
    v2f b0; b0.x = w0[0];  b0.y = w0[DOUT + 0];
    v2f b1; b1.x = w0[16]; b1.y = w0[DOUT + 16];
    v2f b2; b2.x = w0[32]; b2.y = w0[DOUT + 32];
    v2f b3; b3.x = w0[48]; b3.y = w0[DOUT + 48];

    acc0 = __builtin_amdgcn_wmma_f32_16x16x4_f32(false, a, false, b0, (short)0, acc0, false, false);
    acc1 = __builtin_amdgcn_wmma_f32_16x16x4_f32(false, a, false, b1, (short)0, acc1, false, false);
    acc2 = __builtin_amdgcn_wmma_f32_16x16x4_f32(false, a, false, b2, (short)0, acc2, false, false);
    acc3 = __builtin_amdgcn_wmma_f32_16x16x4_f32(false, a, false, b3, (short)0, acc3, false, false);
  }

  const int rbase = m0 + ((lane >= 16) ? 8 : 0);
#pragma unroll
  for (int v = 0; v < 8; ++v) {
    float* hr = h + (size_t)(rbase + v) * DOUT + ncol;
    hr[0]  = acc0[v];
    hr[16] = acc1[v];
    hr[32] = acc2[v];
    hr[48] = acc3[v];
  }
}

// ---------------- edge scatter: out[dst] += h[src] * dinv[src]*dinv[dst] ----------------
// One wave per edge; each lane handles cols {lane, lane+32} (coalesced 128B loads).

__global__ void CRD_scatter(const int* __restrict__ src, const int* __restrict__ dst,
                            const float* __restrict__ h, const float* __restrict__ dinv,
                            float* __restrict__ out) {
  const int wave = threadIdx.x >> 5;
  const int lane = threadIdx.x & 31;
  const int e = blockIdx.x * 8 + wave;
  if (e >= NEDGES) return;

  const int s = src[e];
  const int d = dst[e];
  const float norm = dinv[s] * dinv[d];

  const float* hs = h + (size_t)s * DOUT;
  float* od = out + (size_t)d * DOUT;

  atomicAdd(&od[lane],      hs[lane]      * norm);
  atomicAdd(&od[lane + 32], hs[lane + 32] * norm);
}

// ---------------- finalize: self-loop term + bias + ReLU ----------------

__global__ void CRD_finalize(const float* __restrict__ h, const float* __restrict__ dinv,
                             const float* __restrict__ bias, float* __restrict__ out) {
  int i = blockIdx.x * blockDim.x + threadIdx.x;
  if (i >= NNODES * DOUT) return;
  const int node = i >> 6;
  const int col  = i & 63;
  const float di = dinv[node];
  const float v = out[i] + h[i] * di * di + bias[col];
  out[i] = fmaxf(v, 0.0f);
}

// ---------------- launcher ----------------

extern "C" void kernel_launch(void* const* d_in, const int* in_sizes, int n_in,
                              void* d_out, int out_size, void* d_ws, size_t ws_size,
                              hipStream_t stream) {
  const float* x    = (const float*)d_in[0];      // [50000, 256]
  const int*   edge = (const int*)d_in[1];        // [2, 800000]
  const float* W    = (const float*)d_in[2];      // [256, 64]
  const float* bias = (const float*)d_in[3];      // [64]
  float* out = (float*)d_out;                     // [50000, 64]

  float* h    = (float*)d_ws;                     // 50000*64 floats
  float* deg  = h + (size_t)NNODES * DOUT;        // 50000 floats
  float* dinv = deg + NNODES;                     // 50000 floats

  const int* src = edge;                          // row 0
  const int* dst = edge + NEDGES;                 // row 1

  CRD_zero_out <<<(NNODES * DOUT + 255) / 256, 256, 0, stream>>>(out, NNODES * DOUT);
  CRD_deg_init <<<(NNODES + 255) / 256,        256, 0, stream>>>(deg, NNODES);
  CRD_deg_count<<<(NEDGES + 255) / 256,        256, 0, stream>>>(dst, deg, NEDGES);
  CRD_dinv     <<<(NNODES + 255) / 256,        256, 0, stream>>>(deg, dinv, NNODES);

  // 8 waves/block * 16 rows/wave = 128 rows per block
  CRD_gemm     <<<(NNODES + 127) / 128,        256, 0, stream>>>(x, W, h);

  // 8 edges per block (1 wave per edge)
  CRD_scatter  <<<(NEDGES + 7) / 8,            256, 0, stream>>>(src, dst, h, dinv, out);

  CRD_finalize <<<(NNODES * DOUT + 255) / 256, 256, 0, stream>>>(h, dinv, bias, out);
}